// ExpertLoRA_48404281426529
// MI455X (gfx1250) — compile-verified
//
#include <hip/hip_runtime.h>
#include <hip/hip_bf16.h>

// Problem constants (from the reference)
#define NTOK 8192
#define ZD   1024
#define HD   2048
#define WEXP 64
#define RNK  8
#define SCALE_F 1.0f

typedef __attribute__((ext_vector_type(2))) float v2f;
typedef __attribute__((ext_vector_type(8))) float v8f;

// D = A(16x4 f32) * B(4x16 f32) + C(16x16 f32), wave32, one matrix per wave.
__device__ __forceinline__ v8f wmma_f32(v2f a, v2f b, v8f c) {
    return __builtin_amdgcn_wmma_f32_16x16x4_f32(
        /*neg_a=*/false, a, /*neg_b=*/false, b,
        /*c_mod=*/(short)0, c, /*reuse_a=*/false, /*reuse_b=*/false);
}

// ---------------- bucketing pass ----------------

__global__ void el_zero_kernel(int* counts) {
    if (threadIdx.x < WEXP) counts[threadIdx.x] = 0;
}

__global__ void el_hist_kernel(const int* __restrict__ a_idx, int* counts, int n) {
    int i = blockIdx.x * blockDim.x + threadIdx.x;
    if (i < n) atomicAdd(&counts[a_idx[i]], 1);
}

__global__ void el_scan_kernel(const int* __restrict__ counts, int* offsets, int* cursors) {
    if (threadIdx.x == 0) {
        int acc = 0;
        for (int e = 0; e < WEXP; ++e) {
            offsets[e] = acc;
            cursors[e] = acc;
            acc += counts[e];
        }
    }
}

__global__ void el_scatter_kernel(const int* __restrict__ a_idx, int* cursors,
                                  int* perm, int n) {
    int i = blockIdx.x * blockDim.x + threadIdx.x;
    if (i < n) {
        int p = atomicAdd(&cursors[a_idx[i]], 1);
        perm[p] = i;
    }
}

// ---------------- main fused LoRA tile kernel ----------------
// grid = (64 experts, 512 max tiles), block = one wave32 per 16-token tile.
//
// Note on padding: B fragments for the rank-reductions use (col & 7), so
// B columns 8..15 DUPLICATE ranks 0..7 instead of being zero.  The resulting
// C columns 8..15 are duplicates that are simply never read (we spill only
// cols 0..7), so no zero-masking VALU op is needed on the load->wmma path.

__global__ __launch_bounds__(32)
void el_tile_kernel(const float* __restrict__ z,
                    const float* __restrict__ h_pre,
                    const float* __restrict__ out_pre,
                    const float* __restrict__ a1,   // [W][R][Z]
                    const float* __restrict__ b1,   // [W][H][R]
                    const float* __restrict__ a2,   // [W][R][H]
                    const float* __restrict__ b2,   // [W][Z][R]
                    const int*   __restrict__ counts,
                    const int*   __restrict__ offsets,
                    const int*   __restrict__ perm,
                    float* __restrict__ out) {
    const int e    = blockIdx.x;
    const int tile = blockIdx.y;
    const int cnt  = counts[e];
    if (tile * 16 >= cnt) return;

    const int lane = threadIdx.x;     // 0..31
    const int col  = lane & 15;       // N-index / M-index for fragments
    const int half = lane >> 4;       // lane half selects K pair
    const int koff = half * 2;        // A/B fragment K offset: {0,1} or {2,3}

    __shared__ int   tok[16];
    __shared__ float t1buf[16][RNK];
    __shared__ float hchunk[16][16];
    __shared__ float t2buf[16][RNK];

    int valid = cnt - tile * 16;
    if (valid > 16) valid = 16;

    if (lane < 16) {
        int slot = (lane < valid) ? lane : 0;   // clamp: duplicate rows, never stored
        tok[lane] = perm[offsets[e] + tile * 16 + slot];
    }
    __syncthreads();

    const float* A1e = a1 + (size_t)e * RNK * ZD;   // [R][Z]
    const float* B1e = b1 + (size_t)e * HD * RNK;   // [H][R]
    const float* A2e = a2 + (size_t)e * RNK * HD;   // [R][H]
    const float* B2e = b2 + (size_t)e * ZD * RNK;   // [Z][R]

    // Per-lane cached token rows for the elementwise phases:
    // accumulator layout is row m = v + 8*half, col n = col.
    int myTok[8];
    #pragma unroll
    for (int v = 0; v < 8; ++v) myTok[v] = tok[v + 8 * half];

    // ---------- Phase 1: t1[16 x 8] = z_tile[16 x Z] * A1^T ----------
    const float* zrow  = z   + (size_t)tok[col] * ZD;    // A row m = col
    const float* a1row = A1e + (size_t)(col & 7) * ZD;   // B col n = rank (dup 8..15)
    v8f c1 = {0.f,0.f,0.f,0.f,0.f,0.f,0.f,0.f};
    for (int d0 = 0; d0 < ZD; d0 += 4) {
        v2f a = *(const v2f*)(zrow  + d0 + koff);   // A[m][koff..koff+1]
        v2f b = *(const v2f*)(a1row + d0 + koff);   // B[koff..koff+1][n]
        c1 = wmma_f32(a, b, c1);
    }
    // Spill t1 (cols 0..7 only) to LDS for C->A relayout.
    if (col < RNK) {
        #pragma unroll
        for (int v = 0; v < 8; ++v) t1buf[v + 8 * half][col] = c1[v];
    }
    __syncthreads();

    // Reload t1 in A layout: lane holds row m = col, K pair koff.
    v2f t1lo = { t1buf[col][koff],     t1buf[col][koff + 1] };
    v2f t1hi = { t1buf[col][4 + koff], t1buf[col][4 + koff + 1] };

    // ---------- Phase 2+3 fused: d1 -> silu -> t2, chunked over H ----------
    const float* a2row = A2e + (size_t)(col & 7) * HD;
    const float* hprow[8];
    #pragma unroll
    for (int v = 0; v < 8; ++v) hprow[v] = h_pre + (size_t)myTok[v] * HD + col;

    v8f c2 = {0.f,0.f,0.f,0.f,0.f,0.f,0.f,0.f};    // t2 accumulator [16 tok x ranks]
    for (int h0 = 0; h0 < HD; h0 += 16) {
        // d1 chunk [16 tok x 16 h] = t1[16x8] * B1_chunk^T[8x16], two K=4 WMMAs.
        const float* b1row = B1e + (size_t)(h0 + col) * RNK;  // B col n = h0+col
        if (h0 + 16 < HD)
            __builtin_prefetch(B1e + (size_t)(h0 + 16 + col) * RNK, 0, 0);
        v2f blo = *(const v2f*)(b1row + koff);
        v2f bhi = *(const v2f*)(b1row + 4 + koff);
        v8f d = {0.f,0.f,0.f,0.f,0.f,0.f,0.f,0.f};
        d = wmma_f32(t1lo, blo, d);
        d = wmma_f32(t1hi, bhi, d);
        // h = silu(h_pre + d1) directly on the accumulator layout.
        #pragma unroll
        for (int v = 0; v < 8; ++v) {
            float x = hprow[v][h0] + d[v] * SCALE_F;
            hchunk[v + 8 * half][col] = x / (1.0f + __expf(-x));
        }
        __syncthreads();
        // t2 += h_chunk[16x16] * A2_chunk^T[16x8], four K=4 WMMAs.
        #pragma unroll
        for (int q = 0; q < 4; ++q) {
            v2f a = { hchunk[col][4 * q + koff], hchunk[col][4 * q + koff + 1] };
            v2f b = *(const v2f*)(a2row + h0 + 4 * q + koff);
            c2 = wmma_f32(a, b, c2);
        }
        __syncthreads();
    }

    // ---------- Phase 4: d2 = t2 * B2^T, fused out_pre add ----------
    if (col < RNK) {
        #pragma unroll
        for (int v = 0; v < 8; ++v) t2buf[v + 8 * half][col] = c2[v];
    }
    __syncthreads();
    v2f t2lo = { t2buf[col][koff],     t2buf[col][koff + 1] };
    v2f t2hi = { t2buf[col][4 + koff], t2buf[col][4 + koff + 1] };

    const float* oprow[8];
    float*       orow[8];
    #pragma unroll
    for (int v = 0; v < 8; ++v) {
        oprow[v] = out_pre + (size_t)myTok[v] * ZD + col;
        orow[v]  = out     + (size_t)myTok[v] * ZD + col;
    }

    for (int d0 = 0; d0 < ZD; d0 += 16) {
        const float* b2row = B2e + (size_t)(d0 + col) * RNK;
        v2f blo = *(const v2f*)(b2row + koff);
        v2f bhi = *(const v2f*)(b2row + 4 + koff);
        v8f d = {0.f,0.f,0.f,0.f,0.f,0.f,0.f,0.f};
        d = wmma_f32(t2lo, blo, d);
        d = wmma_f32(t2hi, bhi, d);
        #pragma unroll
        for (int v = 0; v < 8; ++v) {
            if (v + 8 * half < valid) {
                orow[v][d0] = oprow[v][d0] + d[v] * SCALE_F;
            }
        }
    }
}

// ---------------- launch ----------------

extern "C" void kernel_launch(void* const* d_in, const int* in_sizes, int n_in,
                              void* d_out, int out_size, void* d_ws, size_t ws_size,
                              hipStream_t stream) {
    const float* z       = (const float*)d_in[0];
    const int*   a_idx   = (const int*)  d_in[1];
    const float* h_pre   = (const float*)d_in[2];
    const float* out_pre = (const float*)d_in[3];
    const float* a1      = (const float*)d_in[4];
    const float* b1      = (const float*)d_in[5];
    const float* a2      = (const float*)d_in[6];
    const float* b2      = (const float*)d_in[7];
    float* out = (float*)d_out;

    int* counts  = (int*)d_ws;
    int* offsets = counts + WEXP;
    int* cursors = offsets + WEXP;
    int* perm    = cursors + WEXP;

    el_zero_kernel<<<1, WEXP, 0, stream>>>(counts);
    el_hist_kernel<<<NTOK / 256, 256, 0, stream>>>(a_idx, counts, NTOK);
    el_scan_kernel<<<1, 32, 0, stream>>>(counts, offsets, cursors);
    el_scatter_kernel<<<NTOK / 256, 256, 0, stream>>>(a_idx, cursors, perm, NTOK);

    dim3 grid(WEXP, NTOK / 16);   // 64 experts x up-to-512 tiles (early exit)
    el_tile_kernel<<<grid, 32, 0, stream>>>(z, h_pre, out_pre, a1, b1, a2, b2,
                                            counts, offsets, perm, out);
}